// MambaEEG_83416854823100
// MI455X (gfx1250) — compile-verified
//
#include <hip/hip_runtime.h>
#include <math.h>

// Problem constants (from reference)
#define B_      64
#define CIN     22
#define LSEQ    1000
#define DM      22
#define DI      44
#define DCONV   16
#define DTR     2
#define NCLS    4

#define CH      80          // tokens per chunk (5 WMMA row tiles)
#define NT      704         // 44*16 threads = 22 wave32s
#define NWAVES  (NT/32)

typedef float v2f __attribute__((ext_vector_type(2)));
typedef float v8f __attribute__((ext_vector_type(8)));

struct MambaLayerP {
    const float *in_w, *conv_w, *conv_b, *xproj_w, *dt_w, *dt_b, *Alog, *Dp, *out_w, *ln_g, *ln_b;
    int ds;
};

struct Params {
    const float *x, *embed_w, *embed_b, *fc_w, *fc_b;
    MambaLayerP layer[3];
    float* hbuf;   // [B][LSEQ][DM] in global scratch
    float* out;    // [B][NCLS]
};

// ---------------------------------------------------------------------------
// Compile-time-shaped GEMM on V_WMMA_F32_16X16X4_F32.
// All operand loads AND D stores are unguarded (zero-padded LDS buffers), so
// the whole tile lowers to ds_load / v_wmma / ds_store with no exec churn.
// Fragment layouts per CDNA5 ISA 7.12.2:
//  A 16x4 : lanes 0-15 = rows M; VGPR0/1 = K+0/K+1 (lanes 0-15), K+2/K+3 (16-31)
//  B 4x16 : lanes 0-15 = cols N; VGPR0/1 = K+0/K+1 (lanes 0-15), K+2/K+3 (16-31)
//  C/D    : VGPR r = row r (lanes 0-15) / row 8+r (lanes 16-31), lane = col
// ---------------------------------------------------------------------------
template <int MTILES, int NTILES, int KSTEPS, typename LA, typename LB, typename SD>
__device__ __forceinline__ void wmma_gemm(LA loadA, LB loadB, SD storeD)
{
    const int lane = (int)(threadIdx.x & 31);
    const int wave = (int)(threadIdx.x >> 5);
    const int half = lane >> 4;
    const int lp   = lane & 15;
    for (int t = wave; t < MTILES * NTILES; t += NWAVES) {
        const int m0   = (t / NTILES) << 4;
        const int n0   = (t % NTILES) << 4;
        const int mrow = m0 + lp;
        const int ncol = n0 + lp;
        v8f c = {0.f, 0.f, 0.f, 0.f, 0.f, 0.f, 0.f, 0.f};
#pragma unroll
        for (int ki = 0; ki < KSTEPS; ++ki) {
            const int ka = ki * 4 + 2 * half;
            v2f a, b;
            a[0] = loadA(mrow, ka);
            a[1] = loadA(mrow, ka + 1);
            b[0] = loadB(ka, ncol);
            b[1] = loadB(ka + 1, ncol);
            c = __builtin_amdgcn_wmma_f32_16x16x4_f32(false, a, false, b,
                                                      (short)0, c, false, false);
        }
#pragma unroll
        for (int r = 0; r < 8; ++r)
            storeD(m0 + r + 8 * half, ncol, c[r]);
    }
}

__device__ __forceinline__ float silu_f(float v) { return v / (1.f + expf(-v)); }

__global__ __launch_bounds__(NT, 1)
void mamba_eeg_fused(Params p)
{
    const int tid = (int)threadIdx.x;
    const int blk = (int)blockIdx.x;          // batch element

    // LDS (total ~65.2 KB, under the 64 KiB static cap):
    __shared__ float s_big[(CH + 16) * 48];   // conv input (stride 48); rows 16.. alias s_dbl
    __shared__ float s_h[CH * 24];            // staged h chunk (K pad 22->24)
    __shared__ float s_y[CH * 48];            // xc, later y (stride 48)
    __shared__ float s_w_in[24 * 96];         // in_w zero-padded (22x88 -> 24x96)
    __shared__ float s_w_xp[44 * 48];         // xproj_w zero-padded; embed_w during embed
    __shared__ float s_w_out[44 * 32];        // out_w zero-padded (44x22 -> 44x32)
    __shared__ float s_red[2 * NWAVES];
    __shared__ float s_stat[2];
    __shared__ float s_hm[DM];
    __shared__ float s_eb[32];

    float* const s_xpre = s_big;              // stride 48, rows 0..95
    float* const s_dbl  = s_big + 16 * 48;    // stride 48, rows 0..79 (alias rows 16..95)
                                              // also used as GEMM output staging

    float*       hb = p.hbuf + (size_t)blk * LSEQ * DM;
    const float* xb = p.x    + (size_t)blk * CIN * LSEQ;

    // ======== Embedding: h = x^T @ embed_w + embed_b ========
    for (int i = tid; i < 24 * 32; i += NT) {
        const int r = i / 32, cc = i - r * 32;
        s_w_xp[i] = (r < CIN && cc < DM) ? p.embed_w[r * DM + cc] : 0.f;
    }
    if (tid < 32) s_eb[tid] = (tid < DM) ? p.embed_b[tid] : 0.f;
    __syncthreads();
    for (int g0 = 0; g0 < LSEQ; g0 += CH) {
        const int cl = min(CH, LSEQ - g0);
        for (int i = tid; i < CH * 24; i += NT) {
            const int m = i / 24, k = i - m * 24;
            s_h[i] = (m < cl && k < CIN) ? xb[k * LSEQ + g0 + m] : 0.f;
        }
        __syncthreads();
        wmma_gemm<5, 2, 6>(
            [&](int m, int k) { return s_h[m * 24 + k]; },
            [&](int k, int n) { return s_w_xp[k * 32 + n]; },
            [&](int row, int col, float v) { s_dbl[row * 48 + col] = v; });
        __syncthreads();
        // bulk copy-out with bias (coalesced, no exec churn)
        for (int i = tid; i < cl * DM; i += NT) {
            const int m = i / DM, d = i - m * DM;
            hb[(g0 + m) * DM + d] = s_dbl[m * 48 + d] + s_eb[d];
        }
        __syncthreads();
    }
    __threadfence_block();
    __syncthreads();

    // ======== Three Mamba layers ========
    for (int li = 0; li < 3; ++li) {
        const MambaLayerP L = p.layer[li];
        const int ds   = L.ds;
        const int ndbl = DTR + 2 * ds;

        // --- stage zero-padded weights for this layer ---
        for (int i = tid; i < 24 * 96; i += NT) {
            const int r = i / 96, cc = i - r * 96;
            s_w_in[i] = (r < DM && cc < 2 * DI) ? L.in_w[r * (2 * DI) + cc] : 0.f;
        }
        for (int i = tid; i < 44 * 48; i += NT) {
            const int r = i / 48, cc = i - r * 48;
            s_w_xp[i] = (cc < ndbl) ? L.xproj_w[r * ndbl + cc] : 0.f;
        }
        for (int i = tid; i < 44 * 32; i += NT) {
            const int r = i / 32, cc = i - r * 32;
            s_w_out[i] = (cc < DM) ? L.out_w[r * DM + cc] : 0.f;
        }
        // zero causal-conv history rows
        for (int i = tid; i < 16 * 48; i += NT) s_xpre[i] = 0.f;

        // per-thread scan constants (thread owns (d = sd, s = ss))
        const int  sd  = tid / ds;
        const int  ss  = tid - sd * ds;
        const bool act = (tid < DI * ds);
        float c_dtw0 = 0.f, c_dtw1 = 0.f, c_dtb = 0.f, c_A = 0.f, c_D = 0.f;
        if (act) {
            c_dtw0 = L.dt_w[sd];            // dt_w[0][d]
            c_dtw1 = L.dt_w[DI + sd];       // dt_w[1][d]
            c_dtb  = L.dt_b[sd];
            c_A    = -expf(L.Alog[sd * ds + ss]);
            c_D    = L.Dp[sd];
        }
        float hstate = 0.f;
        __syncthreads();

        for (int g0 = 0; g0 < LSEQ; g0 += CH) {
            const int cl = min(CH, LSEQ - g0);

            // P0: stage h chunk (zero-padded) into LDS
            for (int i = tid; i < CH * 24; i += NT) {
                const int m = i / 24, k = i - m * 24;
                s_h[i] = (m < cl && k < DM) ? hb[(g0 + m) * DM + k] : 0.f;
            }
            __syncthreads();

            // P1: xpre = h @ in_w[:, :DI] -> s_xpre rows 15..94 (unguarded LDS store)
            wmma_gemm<5, 3, 6>(
                [&](int m, int k) { return s_h[m * 24 + k]; },
                [&](int k, int n) { return s_w_in[k * 96 + n]; },
                [&](int row, int col, float v) { s_xpre[(row + 15) * 48 + col] = v; });
            __syncthreads();

            // P2: causal depthwise conv (16 taps) + bias + SiLU -> s_y (= xc)
            for (int i = tid; i < cl * DI; i += NT) {
                const int m = i / DI, d = i - m * DI;
                float acc = L.conv_b[d];
#pragma unroll
                for (int j = 0; j < DCONV; ++j)
                    acc += L.conv_w[d * DCONV + j] * s_xpre[(m + j) * 48 + d];
                s_y[m * 48 + d] = silu_f(acc);
            }
            __syncthreads();

            // carry last 15 xpre rows as next chunk's history (before s_dbl aliases)
            for (int i = tid; i < 15 * 48; i += NT) {
                const int r = i / 48, cc = i - r * 48;
                s_xpre[r * 48 + cc] = s_xpre[(r + cl) * 48 + cc];
            }
            __syncthreads();

            // P3: x_dbl = xc @ xproj_w -> s_dbl (stride 48)
            if (ndbl > 16) {
                wmma_gemm<5, 3, 11>(
                    [&](int m, int k) { return s_y[m * 48 + k]; },
                    [&](int k, int n) { return s_w_xp[k * 48 + n]; },
                    [&](int row, int col, float v) { s_dbl[row * 48 + col] = v; });
            } else {
                wmma_gemm<5, 1, 11>(
                    [&](int m, int k) { return s_y[m * 48 + k]; },
                    [&](int k, int n) { return s_w_xp[k * 48 + n]; },
                    [&](int row, int col, float v) { s_dbl[row * 48 + col] = v; });
            }
            __syncthreads();

            // P4: selective scan; thread (d,s) advances cl tokens
            if (act) {
                for (int m = 0; m < cl; ++m) {
                    const float dt0 = s_dbl[m * 48 + 0];
                    const float dt1 = s_dbl[m * 48 + 1];
                    float pre   = dt0 * c_dtw0 + dt1 * c_dtw1 + c_dtb;
                    float delta = (pre > 20.f) ? pre : log1pf(expf(pre));
                    const float xc = s_y[m * 48 + sd];
                    const float Bm = s_dbl[m * 48 + DTR + ss];
                    const float Cm = s_dbl[m * 48 + DTR + ds + ss];
                    hstate = expf(delta * c_A) * hstate + delta * xc * Bm;
                    float contrib = hstate * Cm;
                    for (int off = ds >> 1; off > 0; off >>= 1)
                        contrib += __shfl_xor(contrib, off, ds);
                    if (ss == 0) s_y[m * 48 + sd] = contrib + c_D * xc;  // y + D*xc
                }
            }
            __syncthreads();

            // P5: z = h @ in_w[:, DI:2*DI] -> s_dbl
            wmma_gemm<5, 3, 6>(
                [&](int m, int k) { return s_h[m * 24 + k]; },
                [&](int k, int n) { return s_w_in[k * 96 + DI + n]; },
                [&](int row, int col, float v) { s_dbl[row * 48 + col] = v; });
            __syncthreads();

            // P6: gate y *= silu(z)
            for (int i = tid; i < cl * DI; i += NT) {
                const int m = i / DI, d = i - m * DI;
                s_y[m * 48 + d] *= silu_f(s_dbl[m * 48 + d]);
            }
            __syncthreads();

            // P7: h_new = y @ out_w -> stage in s_dbl (unguarded), then bulk copy
            wmma_gemm<5, 2, 11>(
                [&](int m, int k) { return s_y[m * 48 + k]; },
                [&](int k, int n) { return s_w_out[k * 32 + n]; },
                [&](int row, int col, float v) { s_dbl[row * 48 + col] = v; });
            __syncthreads();
            for (int i = tid; i < cl * DM; i += NT) {
                const int m = i / DM, d = i - m * DM;
                hb[(g0 + m) * DM + d] = s_dbl[m * 48 + d];
            }
            __threadfence_block();
            __syncthreads();
        } // chunk loop

        // ---- LayerNorm over the full (L, DM) slab, per-element affine ----
        {
            float sum = 0.f, sumsq = 0.f;
            for (int i = tid; i < LSEQ * DM; i += NT) {
                const float v = hb[i];
                sum += v; sumsq += v * v;
            }
            for (int off = 16; off > 0; off >>= 1) {
                sum   += __shfl_xor(sum,   off, 32);
                sumsq += __shfl_xor(sumsq, off, 32);
            }
            if ((tid & 31) == 0) {
                s_red[tid >> 5]            = sum;
                s_red[NWAVES + (tid >> 5)] = sumsq;
            }
            __syncthreads();
            if (tid == 0) {
                float a = 0.f, b = 0.f;
                for (int w = 0; w < NWAVES; ++w) { a += s_red[w]; b += s_red[NWAVES + w]; }
                const float n    = (float)(LSEQ * DM);
                const float mean = a / n;
                const float var  = b / n - mean * mean;
                s_stat[0] = mean;
                s_stat[1] = rsqrtf(var + 1e-5f);
            }
            __syncthreads();
            const float mean = s_stat[0], inv = s_stat[1];
            for (int i = tid; i < LSEQ * DM; i += NT)
                hb[i] = (hb[i] - mean) * inv * L.ln_g[i] + L.ln_b[i];
            __threadfence_block();
            __syncthreads();
        }
    } // layer loop

    // ---- Mean over L, then classifier ----
    if (tid < DM) {
        float acc = 0.f;
        for (int l = 0; l < LSEQ; ++l) acc += hb[l * DM + tid];
        s_hm[tid] = acc * (1.f / (float)LSEQ);
    }
    __syncthreads();
    if (tid < NCLS) {
        float acc = p.fc_b[tid];
        for (int dm = 0; dm < DM; ++dm) acc += s_hm[dm] * p.fc_w[dm * NCLS + tid];
        p.out[blk * NCLS + tid] = acc;
    }
}

extern "C" void kernel_launch(void* const* d_in, const int* in_sizes, int n_in,
                              void* d_out, int out_size, void* d_ws, size_t ws_size,
                              hipStream_t stream)
{
    (void)in_sizes; (void)n_in; (void)out_size; (void)ws_size;
    Params p;
    p.x       = (const float*)d_in[0];
    p.embed_w = (const float*)d_in[1];
    p.embed_b = (const float*)d_in[2];
    const int ds_arr[3] = {16, 4, 1};
    for (int i = 0; i < 3; ++i) {
        const int base = 3 + i * 11;
        p.layer[i].in_w    = (const float*)d_in[base + 0];
        p.layer[i].conv_w  = (const float*)d_in[base + 1];
        p.layer[i].conv_b  = (const float*)d_in[base + 2];
        p.layer[i].xproj_w = (const float*)d_in[base + 3];
        p.layer[i].dt_w    = (const float*)d_in[base + 4];
        p.layer[i].dt_b    = (const float*)d_in[base + 5];
        p.layer[i].Alog    = (const float*)d_in[base + 6];
        p.layer[i].Dp      = (const float*)d_in[base + 7];
        p.layer[i].out_w   = (const float*)d_in[base + 8];
        p.layer[i].ln_g    = (const float*)d_in[base + 9];
        p.layer[i].ln_b    = (const float*)d_in[base + 10];
        p.layer[i].ds      = ds_arr[i];
    }
    p.fc_w = (const float*)d_in[36];
    p.fc_b = (const float*)d_in[37];
    p.hbuf = (float*)d_ws;          // 64 * 1000 * 22 floats = 5.6 MB (L2-resident)
    p.out  = (float*)d_out;

    mamba_eeg_fused<<<dim3(B_), dim3(NT), 0, stream>>>(p);
}